// TriPlaneEncoder_72713796321883
// MI455X (gfx1250) — compile-verified
//
#include <hip/hip_runtime.h>

#define FEAT 4
#define PRES 1024
#define GRES 256

typedef float f4 __attribute__((ext_vector_type(4)));

__device__ __forceinline__ f4 ld4(const float* __restrict__ base, int cell) {
    // cell is a texel index; each texel is 4 floats, 16-byte aligned -> global_load_b128
    return *reinterpret_cast<const f4*>(base + (size_t)cell * 4);
}

__device__ __forceinline__ int clampi(int v, int hi) {
    return v < 0 ? 0 : (v > hi ? hi : v);
}

// Bilinear sample of one 1024x1024x4 plane at (pa, pb) in [0,1).
__device__ __forceinline__ f4 bilerp(const float* __restrict__ t, float pa, float pb) {
    float xa = fmaf(pa, (float)(PRES - 1), 0.5f);
    float xb = fmaf(pb, (float)(PRES - 1), 0.5f);
    float fa0 = floorf(xa), fb0 = floorf(xb);
    float fa = xa - fa0,    fb = xb - fb0;
    int a0 = clampi((int)fa0, PRES - 1);
    int b0 = clampi((int)fb0, PRES - 1);
    int a1 = a0 + 1 > PRES - 1 ? PRES - 1 : a0 + 1;
    int b1 = b0 + 1 > PRES - 1 ? PRES - 1 : b0 + 1;

    int r0 = b0 * PRES, r1 = b1 * PRES;
    f4 t00 = ld4(t, r0 + a0);
    f4 t10 = ld4(t, r0 + a1);
    f4 t01 = ld4(t, r1 + a0);
    f4 t11 = ld4(t, r1 + a1);

    float w00 = (1.0f - fa) * (1.0f - fb);
    float w10 = fa * (1.0f - fb);
    float w01 = (1.0f - fa) * fb;
    float w11 = fa * fb;
    return t00 * w00 + t10 * w10 + t01 * w01 + t11 * w11;
}

__global__ __launch_bounds__(256) void triplane_grid_encode_kernel(
    const float* __restrict__ positions,     // (B,3)
    const float* __restrict__ plane,         // (3,1024,1024,4)
    const float* __restrict__ grid,          // (256,256,256,4)
    float* __restrict__ out,                 // (B,16)
    int B)
{
    int i = blockIdx.x * blockDim.x + threadIdx.x;
    if (i >= B) return;

    // Coalesced position read (12B/thread -> global_load_b96)
    const float* p = positions + (size_t)i * 3;
    float px = p[0], py = p[1], pz = p[2];

    // ---- three plane bilerps (planes stay hot in 192MB L2: 48MB footprint) ----
    const float* plane_xy = plane;
    const float* plane_yz = plane + (size_t)PRES * PRES * FEAT;
    const float* plane_zx = plane + (size_t)2 * PRES * PRES * FEAT;
    f4 xy = bilerp(plane_xy, px, py);
    f4 yz = bilerp(plane_yz, py, pz);
    f4 zx = bilerp(plane_zx, pz, px);

    // ---- trilinear grid sample ----
    float gx = fmaf(px, (float)(GRES - 1), 0.5f);
    float gy = fmaf(py, (float)(GRES - 1), 0.5f);
    float gz = fmaf(pz, (float)(GRES - 1), 0.5f);
    float gx0f = floorf(gx), gy0f = floorf(gy), gz0f = floorf(gz);
    float fx = gx - gx0f, fy = gy - gy0f, fz = gz - gz0f;
    int x0 = clampi((int)gx0f, GRES - 1);
    int y0 = clampi((int)gy0f, GRES - 1);
    int z0 = clampi((int)gz0f, GRES - 1);
    int x1 = x0 + 1 > GRES - 1 ? GRES - 1 : x0 + 1;
    int y1 = y0 + 1 > GRES - 1 ? GRES - 1 : y0 + 1;
    int z1 = z0 + 1 > GRES - 1 ? GRES - 1 : z0 + 1;

    float wx0 = 1.0f - fx, wx1 = fx;
    float wy0 = 1.0f - fy, wy1 = fy;
    float wz0 = 1.0f - fz, wz1 = fz;

    int zy00 = (z0 * GRES + y0) * GRES;
    int zy01 = (z0 * GRES + y1) * GRES;
    int zy10 = (z1 * GRES + y0) * GRES;
    int zy11 = (z1 * GRES + y1) * GRES;

    f4 g = ld4(grid, zy00 + x0) * (wx0 * wy0 * wz0);
    g = g + ld4(grid, zy00 + x1) * (wx1 * wy0 * wz0);
    g = g + ld4(grid, zy01 + x0) * (wx0 * wy1 * wz0);
    g = g + ld4(grid, zy01 + x1) * (wx1 * wy1 * wz0);
    g = g + ld4(grid, zy10 + x0) * (wx0 * wy0 * wz1);
    g = g + ld4(grid, zy10 + x1) * (wx1 * wy0 * wz1);
    g = g + ld4(grid, zy11 + x0) * (wx0 * wy1 * wz1);
    g = g + ld4(grid, zy11 + x1) * (wx1 * wy1 * wz1);

    // ---- 64B/thread contiguous output, non-temporal (don't evict grid from L2) ----
    f4* o = reinterpret_cast<f4*>(out + (size_t)i * 16);
    __builtin_nontemporal_store(xy, o + 0);
    __builtin_nontemporal_store(yz, o + 1);
    __builtin_nontemporal_store(zx, o + 2);
    __builtin_nontemporal_store(g,  o + 3);
}

extern "C" void kernel_launch(void* const* d_in, const int* in_sizes, int n_in,
                              void* d_out, int out_size, void* d_ws, size_t ws_size,
                              hipStream_t stream) {
    const float* positions = (const float*)d_in[0];
    const float* plane     = (const float*)d_in[1];
    const float* grid      = (const float*)d_in[2];
    float* out             = (float*)d_out;

    int B = in_sizes[0] / 3;
    dim3 block(256);
    dim3 grd((B + block.x - 1) / block.x);
    triplane_grid_encode_kernel<<<grd, block, 0, stream>>>(positions, plane, grid, out, B);
}